// HopfActivTheta_80221399154900
// MI455X (gfx1250) — compile-verified
//
#include <hip/hip_runtime.h>
#include <math.h>

// ---------------------------------------------------------------------------
// Hopf oscillator Dormand-Prince 5(4) adaptive integrator, fp64, for MI455X.
//
// Shape: B=128, H=256 -> 32768 oscillators, state (r, w) + params (a, b).
// 512 serial adaptive steps; each step needs a GLOBAL RMS error norm over the
// whole state (one shared dt). Latency/sync-bound, not BW/FLOP bound:
//  - state (512 KB fp64) lives entirely in VGPRs (8 osc/thread)
//  - per-step reduction: wave32 shuffle -> LDS -> 1 double/WG to L2
//  - cross-WG sync per step: monotonic atomic-counter barrier (correct on any
//    dispatch) + CDNA5 cluster barrier fast path (s_barrier_signal/wait -3;
//    exact NOP when ClusterID==0 per CDNA5 ISA)
//  - scalar step controller recomputed redundantly (in parallel) per thread
//    from the same 16 partials -> bitwise-identical, no second broadcast.
//
// Precision split: the r-component error (the accept decision-maker) is full
// fp64 (rtol=1.49e-8 demands it). The w-component error is analytically
// h*(sum E)*b ~ 1e-17*b (the E coefficients sum to 0 exactly; only fp64
// rounding residue remains), contributing ~1e-21 per element to a sum with
// accept threshold 65536 — so its divide is done in fp32 (v_rcp_f32 path),
// removing 8 fp64 divide expansions per thread per serial step.
//
// No WMMA: fp64 problem, CDNA5 WMMA has no f64 shape, and RK stages are
// serially dependent. CDNA5 features used instead: cluster barrier, AGENT-
// scoped release/acquire atomics, wave32 shuffles, global_prefetch_b8.
// ---------------------------------------------------------------------------

#define NWG        16          // cluster max = 16 workgroups
#define TPB        256         // 8 wave32 per WG
#define PER_THREAD 8           // 16*256*8 = 32768 oscillators
#define NOSC       32768
#define NSTEPS     512

// Dormand-Prince tableau (fp64)
#define A21 (1.0/5.0)
#define A31 (3.0/40.0)
#define A32 (9.0/40.0)
#define A41 (44.0/45.0)
#define A42 (-56.0/15.0)
#define A43 (32.0/9.0)
#define A51 (19372.0/6561.0)
#define A52 (-25360.0/2187.0)
#define A53 (64448.0/6561.0)
#define A54 (-212.0/729.0)
#define A61 (9017.0/3168.0)
#define A62 (-355.0/33.0)
#define A63 (46732.0/5247.0)
#define A64 (49.0/176.0)
#define A65 (-5103.0/18656.0)
#define B1  (35.0/384.0)
#define B3  (500.0/1113.0)
#define B4  (125.0/192.0)
#define B5  (-2187.0/6784.0)
#define B6  (11.0/84.0)
#define E1  (B1 - 5179.0/57600.0)
#define E3  (B3 - 7571.0/16695.0)
#define E4  (B4 - 393.0/640.0)
#define E5  (B5 + 92097.0/339200.0)
#define E6  (B6 - 187.0/2100.0)
#define E7  (-1.0/40.0)
#define ESUM (E1 + E3 + E4 + E5 + E6 + E7)   // ~1e-17 (fp64 rounding residue)

#define RTOL  1.49012e-08
#define ATOL  1.49012e-08
#define RTOLF 1.49012e-08f
#define ATOLF 1.49012e-08f
#define T1_   6.283185307179586476925286766559  // 2*pi
#define DT0   0.05

__device__ __forceinline__ double fr(double y, double a) {
    return (a - y * y) * y;   // dr = (a - r^2) r
}

__global__ __launch_bounds__(TPB)
void hopf_dopri_cluster(const float* __restrict__ r_in,
                        const float* __restrict__ w_in,
                        const float* __restrict__ a_in,
                        const float* __restrict__ b_in,
                        float* __restrict__ out,
                        double* __restrict__ partials,     // [2][NWG]
                        unsigned int* __restrict__ counter)
{
    const int tid  = threadIdx.x;
    const int wg   = blockIdx.x;
    const int lane = tid & 31;      // wave32
    const int wave = tid >> 5;

    __shared__ double s_wave[TPB / 32];

    // ---- load state/params into registers (one-time, ~0.5 MB total) ----
    const int e0 = wg * (TPB * PER_THREAD) + tid;
    // gfx1250 prefetch path (global_prefetch_b8)
    __builtin_prefetch(r_in + e0, 0, 1);
    __builtin_prefetch(w_in + e0, 0, 1);
    __builtin_prefetch(a_in + e0, 0, 1);
    __builtin_prefetch(b_in + e0, 0, 1);

    double r[PER_THREAD], w[PER_THREAD], a[PER_THREAD], b[PER_THREAD];
    float  bf[PER_THREAD];
#pragma unroll
    for (int i = 0; i < PER_THREAD; ++i) {
        const int e = e0 + i * TPB;
        r[i]  = (double)r_in[e];
        w[i]  = (double)w_in[e];
        a[i]  = (double)a_in[e];
        bf[i] = b_in[e];
        b[i]  = (double)bf[i];
    }

    double t  = 0.0;
    double dt = DT0;

    for (int step = 0; step < NSTEPS; ++step) {
        const bool   done = (t >= T1_ * (1.0 - 1e-12));
        const double h    = fmin(dt, T1_ - t);
        const float  hEf  = (float)h * (float)ESUM;   // fp32 w-error scale

        double y5r[PER_THREAD], w5v[PER_THREAD];
        double local  = 0.0;   // fp64: r-component error norm terms
        float  localw = 0.0f;  // fp32: negligible w-component terms

        // ---- full RK45 step per oscillator, all in registers ----
#pragma unroll
        for (int i = 0; i < PER_THREAD; ++i) {
            const double r0 = r[i], ai = a[i];
            const double k1 = fr(r0, ai);
            const double y2 = fma(h, A21 * k1, r0);
            const double k2 = fr(y2, ai);
            const double s3 = fma(A32, k2, A31 * k1);
            const double y3 = fma(h, s3, r0);
            const double k3 = fr(y3, ai);
            const double s4 = fma(A43, k3, fma(A42, k2, A41 * k1));
            const double y4 = fma(h, s4, r0);
            const double k4 = fr(y4, ai);
            const double s5 = fma(A54, k4, fma(A53, k3, fma(A52, k2, A51 * k1)));
            const double y5s = fma(h, s5, r0);
            const double k5 = fr(y5s, ai);
            const double s6 = fma(A65, k5, fma(A64, k4,
                               fma(A63, k3, fma(A62, k2, A61 * k1))));
            const double y6 = fma(h, s6, r0);
            const double k6 = fr(y6, ai);
            const double sb = fma(B6, k6, fma(B5, k5,
                               fma(B4, k4, fma(B3, k3, B1 * k1))));
            const double y5 = fma(h, sb, r0);
            const double k7 = fr(y5, ai);
            const double se = fma(E7, k7, fma(E6, k6, fma(E5, k5,
                               fma(E4, k4, fma(E3, k3, E1 * k1)))));
            const double err = h * se;
            const double sc  = fma(RTOL, fmax(fabs(r0), fabs(y5)), ATOL);
            const double q   = err / sc;          // fp64 divide: decision-maker
            local = fma(q, q, local);
            y5r[i] = y5;

            // w component: all RK stages equal b -> w5 = w + h*b exactly.
            // Error term is ~1e-17*b -> contribution ~1e-21; fp32 suffices.
            const double w0v = w[i];
            const double w5  = fma(h, b[i], w0v);
            const float  errw = hEf * bf[i];
            const float  scw  = fmaf(RTOLF,
                                     fmaxf(fabsf((float)w0v), fabsf((float)w5)),
                                     ATOLF);
            const float  qw   = errw / scw;       // fp32 v_rcp_f32 path
            localw = fmaf(qw, qw, localw);
            w5v[i] = w5;
        }
        local += (double)localw;

        // ---- wave32 shuffle reduction ----
#pragma unroll
        for (int off = 16; off > 0; off >>= 1)
            local += __shfl_down(local, off, 32);
        if (lane == 0) s_wave[wave] = local;
        __syncthreads();

        // ---- one partial per WG to L2 (double-buffered by step parity) ----
        double* pbuf = partials + (step & 1) * NWG;
        if (tid == 0) {
            double wgsum = 0.0;
#pragma unroll
            for (int j = 0; j < TPB / 32; ++j) wgsum += s_wave[j];
            __hip_atomic_store(&pbuf[wg], wgsum,
                               __ATOMIC_RELEASE, __HIP_MEMORY_SCOPE_AGENT);
            __hip_atomic_fetch_add(counter, 1u,
                               __ATOMIC_RELEASE, __HIP_MEMORY_SCOPE_AGENT);
        }

        // ---- cross-WG barrier ----
        // Fast path: CDNA5 cluster barrier (s_barrier_signal -3 / wait -3),
        // one wave per WG per ISA recommendation; exact NOP when not
        // dispatched as a cluster. Correctness carried by the monotonic
        // counter spin regardless of cluster support.
        if (wave == 0) __builtin_amdgcn_s_cluster_barrier();
        if (tid == 0) {
            const unsigned int target = (unsigned int)(step + 1) * NWG;
            while (__hip_atomic_load(counter, __ATOMIC_ACQUIRE,
                                     __HIP_MEMORY_SCOPE_AGENT) < target)
                __builtin_amdgcn_s_sleep(2);
        }
        __syncthreads();

        // ---- every thread rebuilds the identical scalar controller ----
        double total = 0.0;
#pragma unroll
        for (int j = 0; j < NWG; ++j)
            total += __hip_atomic_load(&pbuf[j], __ATOMIC_RELAXED,
                                       __HIP_MEMORY_SCOPE_AGENT);
        const double enorm  = sqrt(total / (double)(2 * NOSC));
        const bool   accept = (enorm <= 1.0) && !done;

        if (accept) {
#pragma unroll
            for (int i = 0; i < PER_THREAD; ++i) { r[i] = y5r[i]; w[i] = w5v[i]; }
            t = t + h;
        }
        const double safe = fmax(enorm, 1e-10);
        const double fac  = fmin(fmax(0.9 * pow(safe, -0.2), 0.2), 10.0);
        dt = done ? dt : h * fac;
        __syncthreads();   // protect s_wave & pbuf reuse next iteration
    }

    // ---- write outputs: [r (32768 f32) | w (32768 f32)] ----
#pragma unroll
    for (int i = 0; i < PER_THREAD; ++i) {
        const int e = e0 + i * TPB;
        out[e]        = (float)r[i];
        out[NOSC + e] = (float)w[i];
    }
}

extern "C" void kernel_launch(void* const* d_in, const int* in_sizes, int n_in,
                              void* d_out, int out_size, void* d_ws, size_t ws_size,
                              hipStream_t stream) {
    // setup_inputs order: stp(int,1), r, w, a, b  (each 128*256 f32)
    const float* r = (const float*)d_in[1];
    const float* w = (const float*)d_in[2];
    const float* a = (const float*)d_in[3];
    const float* b = (const float*)d_in[4];
    float* out = (float*)d_out;

    // d_ws layout: [0..3] counter (u32), [128..383] partials double[2][16]
    unsigned int* counter = (unsigned int*)d_ws;
    double* partials = (double*)((char*)d_ws + 128);

    // counter must start at 0 each launch; memset is graph-capturable
    hipMemsetAsync(d_ws, 0, 128, stream);

    hipLaunchKernelGGL(hopf_dopri_cluster, dim3(NWG), dim3(TPB), 0, stream,
                       r, w, a, b, out, partials, counter);
}